// LSTMWithMaxXBar_69191923139127
// MI455X (gfx1250) — compile-verified
//
#include <hip/hip_runtime.h>

#define B_  32
#define T_  512
#define D_  256
#define HS_ 512
#define G_  2048   // 4*HS
#define NBLK_SCAN 16

typedef float v2f __attribute__((ext_vector_type(2)));
typedef float v8f __attribute__((ext_vector_type(8)));
typedef int   v4i __attribute__((ext_vector_type(4)));

// ---- optional CDNA5 async global->LDS path (guarded; falls back to ld/st) ----
#if defined(__has_builtin)
#if __has_builtin(__builtin_amdgcn_global_load_async_to_lds_b128)
#define USE_ASYNC_LDS 1
#endif
#endif

#ifdef USE_ASYNC_LDS
#if __has_builtin(__builtin_amdgcn_s_wait_asynccnt)
#define WAIT_ASYNCCNT0() __builtin_amdgcn_s_wait_asynccnt(0)
#else
#define WAIT_ASYNCCNT0() asm volatile("s_wait_asynccnt 0x0" ::: "memory")
#endif
// signature (from hipcc diagnostic): (v4i AS1*, v4i AS3*, imm offset, imm cpol)
#define ASYNC_B128(gsrc, ldst)                                              \
    __builtin_amdgcn_global_load_async_to_lds_b128(                         \
        (__attribute__((address_space(1))) v4i*)(gsrc),                     \
        (__attribute__((address_space(3))) v4i*)(ldst), 0, 0)
#endif

__device__ __forceinline__ v8f wmma_f32(v2f a, v2f b, v8f c) {
    // D = A(16x4) * B(4x16) + C(16x16), all f32, wave32
    return __builtin_amdgcn_wmma_f32_16x16x4_f32(false, a, false, b, (short)0, c, false, false);
}

__device__ __forceinline__ float sigmoid_(float x) {
    return 1.0f / (1.0f + __expf(-x));
}
__device__ __forceinline__ float tanh_(float x) {
    // stable at +/-inf: tanh(x) = 1 - 2/(exp(2x)+1)
    return 1.0f - 2.0f / (__expf(2.0f * x) + 1.0f);
}

// ---------------------------------------------------------------------------
// Kernel 1: xbar = shifted cummax of x along t; zero h ping-pong buf; reset bar
// grid: 32 x 256  (B*D = 8192 threads), thread -> (b, d)
// ---------------------------------------------------------------------------
__global__ __launch_bounds__(256) void prep_kernel(const float* __restrict__ x,
                                                   float* __restrict__ xbar,
                                                   float* __restrict__ hbuf,
                                                   int* __restrict__ bar) {
    const int tid = blockIdx.x * blockDim.x + threadIdx.x;  // 0..8191
    if (tid == 0) *bar = 0;   // barrier counter must be re-armed every call
    hbuf[tid] = 0.0f;         // zero hbuf[0] (B*HS = 16384 floats)
    hbuf[tid + 8192] = 0.0f;

    const int b = tid / D_;
    const int d = tid % D_;
    const float* xp = x + (size_t)b * T_ * D_ + d;
    float* xb = xbar + (size_t)b * T_ * D_ + d;

    float run = xp[0];
    xb[0]  = 0.0f;
    xb[D_] = 0.0f;
    for (int t = 2; t < T_; ++t) {
        xb[t * D_] = run;                        // max over x[0..t-2]
        run = fmaxf(run, xp[(t - 1) * D_]);
    }
}

// ---------------------------------------------------------------------------
// Kernel 2: gx[m, n] = x@W_ih + xbar@W_iBarh + bias, m = t*B + b (t-major)
// grid: (128, 32), block 256 (8 waves). Block tile 128(M) x 64(N).
// W panel (256K x 64N = 64 KB) async-staged in LDS once per pass (8-wave reuse).
// ---------------------------------------------------------------------------
__global__ __launch_bounds__(256) void gx_gemm_kernel(const float* __restrict__ x,
                                                      const float* __restrict__ xbar,
                                                      const float* __restrict__ w_ih,
                                                      const float* __restrict__ w_ibh,
                                                      const float* __restrict__ bias,
                                                      float* __restrict__ gx) {
    __shared__ float wtile[D_ * 64];   // [k][n] row-major, 64 KB

    const int lane = threadIdx.x & 31;
    const int wave = threadIdx.x >> 5;
    const int lo = lane & 15;
    const int hi = lane >> 4;

    const int m0 = blockIdx.x * 128 + wave * 16;
    const int n0 = blockIdx.y * 64;

    // A row carried by this lane: row (m0+lo); map m -> (t, b)
    const int mT = m0 + lo;
    const int tA = mT >> 5;      // /B
    const int bA = mT & 31;      // %B
    const float* xrow  = x    + (size_t)bA * T_ * D_ + (size_t)tA * D_;
    const float* xbrow = xbar + (size_t)bA * T_ * D_ + (size_t)tA * D_;

    const v8f vzero = {0.f,0.f,0.f,0.f,0.f,0.f,0.f,0.f};
    v8f acc[4] = {vzero, vzero, vzero, vzero};

    for (int pass = 0; pass < 2; ++pass) {
        const float* w    = pass ? w_ibh : w_ih;
        const float* arow = pass ? xbrow : xrow;

        __syncthreads();   // previous pass done reading LDS
#ifdef USE_ASYNC_LDS
        if (lane < 16) {   // 16 lanes x b128 cover one 64-float W row
            for (int k = wave; k < D_; k += 8) {
                ASYNC_B128(w + (size_t)k * G_ + n0 + lane * 4,
                           &wtile[k * 64 + lane * 4]);
            }
        }
        WAIT_ASYNCCNT0();
#else
        for (int idx = threadIdx.x; idx < D_ * 64; idx += 256) {
            wtile[idx] = w[(size_t)(idx >> 6) * G_ + n0 + (idx & 63)];
        }
#endif
        __syncthreads();

        #pragma unroll 4
        for (int k = 0; k < D_; k += 4) {
            v2f a = *(const v2f*)(arow + k + 2 * hi);   // A[row, k+2hi + {0,1}]
            #pragma unroll
            for (int j = 0; j < 4; ++j) {
                v2f bf;
                bf[0] = wtile[(k + 2 * hi)     * 64 + 16 * j + lo];
                bf[1] = wtile[(k + 2 * hi + 1) * 64 + 16 * j + lo];
                acc[j] = wmma_f32(a, bf, acc[j]);
            }
        }
    }

    #pragma unroll
    for (int j = 0; j < 4; ++j) {
        const int n = n0 + 16 * j + lo;
        const float bv = bias[n];
        #pragma unroll
        for (int r = 0; r < 8; ++r) {
            const int row = m0 + r + 8 * hi;    // C layout: VGPR r -> M=r (+8 for hi lanes)
            gx[(size_t)row * G_ + n] = acc[j][r] + bv;
        }
    }
}

// ---------------------------------------------------------------------------
// Kernel 3: sequential LSTM scan. Persistent: 16 blocks x 128 threads (64 waves).
// Block owns 32 hs columns. W_hh slice (512K x 128c) staged ONCE in LDS using a
// K-pair-interleaved layout so B-frags are single ds_load_b64 (no repack movs).
// K=512 is split across 2 waves per tile (4 SIMDs busy); partial sums reduced
// through a 16 KB LDS scratch. Cell state c stays in kh=0 waves' registers.
// ---------------------------------------------------------------------------
__global__ __launch_bounds__(128) void scan_kernel(const float* __restrict__ gx,
                                                   const float* __restrict__ w_hh,
                                                   float* __restrict__ hbuf,
                                                   int* bar,
                                                   float* __restrict__ out) {
    extern __shared__ float wlds[];   // [256 k2][128 c] float2 pairs = 256 KB, + 16 KB scratch

    const int lane = threadIdx.x & 31;
    const int wave = threadIdx.x >> 5;            // 0..3
    const int colw = wave >> 1;                   // 0..1 : 16-col chunk
    const int kh   = wave & 1;                    // 0..1 : K half
    const int lo = lane & 15;
    const int hi = lane >> 4;
    const int hsB = blockIdx.x * 32;              // block's hs column base
    const int n_base = hsB + colw * 16 + lo;      // global hs column of this lane
    const int c_base = colw * 16 + lo;            // column within LDS row (pre-gate)
    const int k0 = kh * (HS_ / 2);                // this wave's K range

    v2f* wp2 = (v2f*)wlds;                        // pair entries [k2*128 + c]
    float* sred = wlds + HS_ * 128;               // reduction scratch (4096 floats)

    // ---- one-time fill: pair-interleaved W_hh slice {W[2k2][c], W[2k2+1][c]} ----
    for (int idx = threadIdx.x; idx < (HS_ / 2) * 128; idx += 128) {
        const int k2 = idx >> 7;                  // 0..255
        const int c  = idx & 127;
        const float* src = w_hh + (size_t)(2 * k2) * G_ + (c >> 5) * HS_ + hsB + (c & 31);
        v2f p;
        p[0] = src[0];
        p[1] = src[G_];
        wp2[idx] = p;                             // ds_store_b64
    }
    __syncthreads();

    const v8f vzero = {0.f,0.f,0.f,0.f,0.f,0.f,0.f,0.f};
    v8f cstate[2] = {vzero, vzero};               // persistent cell state (kh==0 waves)

    const size_t OUT_H = (size_t)B_ * T_ * HS_;
    const size_t OUT_C = OUT_H + B_ * HS_;

    for (int t = 0; t < T_; ++t) {
        const float* hread  = hbuf + (size_t)(t & 1) * (B_ * HS_);
        float*       hwrite = hbuf + (size_t)((t + 1) & 1) * (B_ * HS_);

        // A) accumulator init: gx tile for kh==0, zero for kh==1
        v8f acc[2][4];
        if (kh == 0) {
            #pragma unroll
            for (int mt = 0; mt < 2; ++mt) {
                #pragma unroll
                for (int gi = 0; gi < 4; ++gi) {
                    v8f ci;
                    #pragma unroll
                    for (int r = 0; r < 8; ++r) {
                        const int b = mt * 16 + r + 8 * hi;
                        ci[r] = gx[(size_t)(t * B_ + b) * G_ + gi * HS_ + n_base];
                    }
                    acc[mt][gi] = ci;
                }
            }
        } else {
            #pragma unroll
            for (int mt = 0; mt < 2; ++mt)
                #pragma unroll
                for (int gi = 0; gi < 4; ++gi) acc[mt][gi] = vzero;
        }

        // B) gates += h_{t-1} @ W_hh over this wave's K half (A global, B LDS b64)
        #pragma unroll 2
        for (int k = k0; k < k0 + HS_ / 2; k += 4) {
            v2f a0 = *(const v2f*)(hread + (size_t)(lo)      * HS_ + k + 2 * hi);
            v2f a1 = *(const v2f*)(hread + (size_t)(16 + lo) * HS_ + k + 2 * hi);
            const int k2h = (k >> 1) + hi;
            #pragma unroll
            for (int gi = 0; gi < 4; ++gi) {
                v2f bf = wp2[k2h * 128 + gi * 32 + c_base];   // single ds_load_b64
                acc[0][gi] = wmma_f32(a0, bf, acc[0][gi]);
                acc[1][gi] = wmma_f32(a1, bf, acc[1][gi]);
            }
        }

        // C) kh==1 waves publish partial sums
        if (kh == 1) {
            #pragma unroll
            for (int mt = 0; mt < 2; ++mt)
                #pragma unroll
                for (int gi = 0; gi < 4; ++gi) {
                    const int slot = colw * 8 + mt * 4 + gi;
                    *(v8f*)(sred + (size_t)slot * 256 + lane * 8) = acc[mt][gi];
                }
        }
        __syncthreads();   // D) partials visible

        // E) kh==0 waves reduce + pointwise LSTM cell update
        if (kh == 0) {
            #pragma unroll
            for (int mt = 0; mt < 2; ++mt) {
                #pragma unroll
                for (int gi = 0; gi < 4; ++gi) {
                    const int slot = colw * 8 + mt * 4 + gi;
                    acc[mt][gi] += *(const v8f*)(sred + (size_t)slot * 256 + lane * 8);
                }
            }
            #pragma unroll
            for (int mt = 0; mt < 2; ++mt) {
                #pragma unroll
                for (int r = 0; r < 8; ++r) {
                    const float ig = sigmoid_(acc[mt][0][r]);
                    const float fg = sigmoid_(acc[mt][1][r]);
                    const float gg = tanh_(acc[mt][2][r]);
                    const float og = sigmoid_(acc[mt][3][r]);
                    const float cv = fg * cstate[mt][r] + ig * gg;
                    cstate[mt][r] = cv;
                    const float hv = og * tanh_(cv);
                    const int b = mt * 16 + r + 8 * hi;
                    hwrite[(size_t)b * HS_ + n_base] = hv;
                    out[(size_t)b * T_ * HS_ + (size_t)t * HS_ + n_base] = hv;
                    if (t == T_ - 1) {
                        out[OUT_H + (size_t)b * HS_ + n_base] = hv;
                        out[OUT_C + (size_t)b * HS_ + n_base] = cv;
                    }
                }
            }
        }

        // F) device-wide step barrier (also provides WAR safety for sred)
        __builtin_amdgcn_fence(__ATOMIC_RELEASE, "agent");   // flush h stores
        __syncthreads();
        if (threadIdx.x == 0) {
            __hip_atomic_fetch_add(bar, 1, __ATOMIC_RELAXED, __HIP_MEMORY_SCOPE_AGENT);
            const int target = NBLK_SCAN * (t + 1);
            while (__hip_atomic_load(bar, __ATOMIC_RELAXED, __HIP_MEMORY_SCOPE_AGENT) < target) {
                __builtin_amdgcn_s_sleep(1);
            }
        }
        __syncthreads();
        __builtin_amdgcn_fence(__ATOMIC_ACQUIRE, "agent");   // invalidate L0
    }
}

// ---------------------------------------------------------------------------
extern "C" void kernel_launch(void* const* d_in, const int* in_sizes, int n_in,
                              void* d_out, int out_size, void* d_ws, size_t ws_size,
                              hipStream_t stream) {
    const float* x     = (const float*)d_in[0];   // (B,T,D)
    const float* w_ih  = (const float*)d_in[1];   // (D,4HS)
    const float* w_ibh = (const float*)d_in[2];   // (D,4HS)
    const float* w_hh  = (const float*)d_in[3];   // (HS,4HS)
    const float* bias  = (const float*)d_in[4];   // (4HS,)
    float* out = (float*)d_out;

    // workspace layout (floats)
    float* ws_f  = (float*)d_ws;
    int*   bar   = (int*)d_ws;                    // [0..15] reserved
    float* hbuf  = ws_f + 16;                     // 2 * B*HS ping-pong
    float* xbar  = hbuf + 2 * (B_ * HS_);         // B*T*D
    float* gx    = xbar + (size_t)B_ * T_ * D_;   // (T*B, 4HS), t-major rows

    prep_kernel<<<dim3((B_ * D_) / 256), dim3(256), 0, stream>>>(x, xbar, hbuf, bar);

    gx_gemm_kernel<<<dim3((B_ * T_) / 128, G_ / 64), dim3(256), 0, stream>>>(
        x, xbar, w_ih, w_ibh, bias, gx);

    const size_t scan_lds = (HS_ * 128 + 16 * 32 * 8) * sizeof(float);  // 272 KB
    scan_kernel<<<dim3(NBLK_SCAN), dim3(128), scan_lds, stream>>>(
        gx, w_hh, hbuf, bar, out);
}